// MambaDC_9740985827498
// MI455X (gfx1250) — compile-verified
//
#include <hip/hip_runtime.h>
#include <hip/hip_bf16.h>
#include <math.h>

// ---------------- model constants ----------------
#define KB    2        // batch
#define KT    1000     // seq len
#define K_IN  257
#define K_OUT 514
#define K_DM  512      // d_model
#define K_DI  1024     // d_inner
#define K_DS  16       // d_state
#define K_DTR 32       // dt_rank
#define K_NL  4        // layers
#define K_BT  (KB*KT)  // 2000 rows

typedef float v2f __attribute__((ext_vector_type(2)));
typedef float v8f __attribute__((ext_vector_type(8)));

#define FLAG_BIAS     1
#define FLAG_SOFTPLUS 2
#define FLAG_RESID    4

__device__ __forceinline__ float silu_f(float v) { return v / (1.0f + __expf(-v)); }
__device__ __forceinline__ float softplus_f(float v) {
    return (v > 20.0f) ? v : logf(1.0f + expf(v));
}

__device__ __forceinline__ v8f wmma4(v2f a, v2f b, v8f c) {
    return __builtin_amdgcn_wmma_f32_16x16x4_f32(
        false, a, false, b, (short)0, c, false, false);
}

// ---------------------------------------------------------------------------
// WMMA f32 GEMM:  C[M,N] = epilogue( A[M,K] @ W[K,N] )
// one wave per 16x64 C strip: 4 accumulators share one A fragment per k-step.
// V_WMMA_F32_16X16X4_F32 fragment layout:
//   A: lane l holds A[m0 + (l&15)][k0 + 2*(l>>4) + j], j=0,1
//   B: lane l holds W[k0 + 2*(l>>4) + j][n0 + (l&15)]
//   C/D: reg r, lane l -> M = m0 + 8*(l>>4) + r, N = n0 + (l&15)
// Fast path (tile fully in-bounds): unconditional loads, 2-stage pipeline.
// ---------------------------------------------------------------------------
__global__ void k_gemm_wmma_f32(const float* __restrict__ A,
                                const float* __restrict__ W,
                                const float* __restrict__ bias,
                                const float* __restrict__ resid,
                                float* __restrict__ C,
                                int M, int N, int K,
                                int lda, int ldw, int ldc, int flags)
{
    const int lane = threadIdx.x & 31;
    const int wave = threadIdx.x >> 5;
    const int tile = blockIdx.x * (blockDim.x >> 5) + wave;

    const int mtiles   = (M + 15) >> 4;
    const int ntiles64 = (N + 63) >> 6;
    if (tile >= mtiles * ntiles64) return;        // whole wave exits together
    const int m0 = (tile / ntiles64) << 4;
    const int n0 = (tile % ntiles64) << 6;

    const int lm = lane & 15;
    const int lk = (lane >> 4) << 1;              // k sub-offset: 0 or 2

    v8f acc0 = {}, acc1 = {}, acc2 = {}, acc3 = {};
    const int K4 = K & ~3;

#define GUARDED_STEP(KA0, KA1)                                              \
    {                                                                       \
        const int am = m0 + lm;                                             \
        const bool am_ok = (am < M);                                        \
        v2f af;                                                             \
        af.x = (am_ok && (KA0) < K) ? A[(size_t)am * lda + (KA0)] : 0.0f;   \
        af.y = (am_ok && (KA1) < K) ? A[(size_t)am * lda + (KA1)] : 0.0f;   \
        {                                                                   \
            const int bn = n0 + 0 * 16 + lm;                                \
            v2f bf;                                                         \
            bf.x = (bn < N && (KA0) < K) ? W[(size_t)(KA0) * ldw + bn] : 0.0f; \
            bf.y = (bn < N && (KA1) < K) ? W[(size_t)(KA1) * ldw + bn] : 0.0f; \
            acc0 = wmma4(af, bf, acc0);                                     \
        }                                                                   \
        {                                                                   \
            const int bn = n0 + 1 * 16 + lm;                                \
            v2f bf;                                                         \
            bf.x = (bn < N && (KA0) < K) ? W[(size_t)(KA0) * ldw + bn] : 0.0f; \
            bf.y = (bn < N && (KA1) < K) ? W[(size_t)(KA1) * ldw + bn] : 0.0f; \
            acc1 = wmma4(af, bf, acc1);                                     \
        }                                                                   \
        {                                                                   \
            const int bn = n0 + 2 * 16 + lm;                                \
            v2f bf;                                                         \
            bf.x = (bn < N && (KA0) < K) ? W[(size_t)(KA0) * ldw + bn] : 0.0f; \
            bf.y = (bn < N && (KA1) < K) ? W[(size_t)(KA1) * ldw + bn] : 0.0f; \
            acc2 = wmma4(af, bf, acc2);                                     \
        }                                                                   \
        {                                                                   \
            const int bn = n0 + 3 * 16 + lm;                                \
            v2f bf;                                                         \
            bf.x = (bn < N && (KA0) < K) ? W[(size_t)(KA0) * ldw + bn] : 0.0f; \
            bf.y = (bn < N && (KA1) < K) ? W[(size_t)(KA1) * ldw + bn] : 0.0f; \
            acc3 = wmma4(af, bf, acc3);                                     \
        }                                                                   \
    }

    if (m0 + 16 <= M && n0 + 64 <= N) {
        // ---------------- fast path: no per-element guards ----------------
        const float* pa = A + (size_t)(m0 + lm) * lda + lk;
        const float* pb = W + (size_t)lk * ldw + n0 + lm;
        const size_t bstep = (size_t)4 * ldw;
        if (K4 > 0) {
            v2f af, b0, b1, b2, b3;
            af.x = pa[0];  af.y = pa[1];
            b0.x = pb[0];  b0.y = pb[ldw];
            b1.x = pb[16]; b1.y = pb[ldw + 16];
            b2.x = pb[32]; b2.y = pb[ldw + 32];
            b3.x = pb[48]; b3.y = pb[ldw + 48];
            for (int k0 = 4; k0 < K4; k0 += 4) {
                pa += 4; pb += bstep;
                v2f afn, b0n, b1n, b2n, b3n;          // prefetch next k-step
                afn.x = pa[0];  afn.y = pa[1];
                b0n.x = pb[0];  b0n.y = pb[ldw];
                b1n.x = pb[16]; b1n.y = pb[ldw + 16];
                b2n.x = pb[32]; b2n.y = pb[ldw + 32];
                b3n.x = pb[48]; b3n.y = pb[ldw + 48];
                acc0 = wmma4(af, b0, acc0);
                acc1 = wmma4(af, b1, acc1);
                acc2 = wmma4(af, b2, acc2);
                acc3 = wmma4(af, b3, acc3);
                af = afn; b0 = b0n; b1 = b1n; b2 = b2n; b3 = b3n;
            }
            acc0 = wmma4(af, b0, acc0);
            acc1 = wmma4(af, b1, acc1);
            acc2 = wmma4(af, b2, acc2);
            acc3 = wmma4(af, b3, acc3);
        }
        if (K4 < K) {                 // K remainder (stem K=257)
            const int ka0 = K4 + lk, ka1 = ka0 + 1;
            GUARDED_STEP(ka0, ka1)
        }
    } else {
        // ---------------- guarded path (head GEMM N-edge tile) ------------
        for (int k0 = 0; k0 < K; k0 += 4) {
            const int ka0 = k0 + lk, ka1 = ka0 + 1;
            GUARDED_STEP(ka0, ka1)
        }
    }
#undef GUARDED_STEP

    // ---------------- epilogue ----------------
    const int mb = m0 + ((lane >> 4) << 3);
#define EPILOG(ACC, S)                                                      \
    {                                                                       \
        const int cn = n0 + (S) * 16 + lm;                                  \
        if (cn < N) {                                                       \
            const float bv = (flags & FLAG_BIAS) ? bias[cn] : 0.0f;         \
            _Pragma("unroll")                                               \
            for (int r = 0; r < 8; ++r) {                                   \
                const int cm = mb + r;                                      \
                if (cm < M) {                                               \
                    float v = (ACC)[r] + bv;                                \
                    if (flags & FLAG_SOFTPLUS) v = softplus_f(v);           \
                    if (flags & FLAG_RESID)    v += resid[(size_t)cm * ldc + cn]; \
                    C[(size_t)cm * ldc + cn] = v;                           \
                }                                                           \
            }                                                               \
        }                                                                   \
    }
    EPILOG(acc0, 0)
    EPILOG(acc1, 1)
    EPILOG(acc2, 2)
    EPILOG(acc3, 3)
#undef EPILOG
}

// ---------------------------------------------------------------------------
// LayerNorm over last dim D (one block per row), optional ReLU
// ---------------------------------------------------------------------------
__global__ void k_layernorm(const float* __restrict__ X,
                            const float* __restrict__ g,
                            const float* __restrict__ b,
                            float* __restrict__ Y, int D, int do_relu)
{
    const int row = blockIdx.x;
    const float* x = X + (size_t)row * D;
    float* y = Y + (size_t)row * D;

    float s = 0.0f, s2 = 0.0f;
    for (int i = threadIdx.x; i < D; i += blockDim.x) {
        float v = x[i];
        s += v; s2 += v * v;
    }
#pragma unroll
    for (int o = 16; o > 0; o >>= 1) {          // wave32 reduction
        s  += __shfl_xor(s,  o, 32);
        s2 += __shfl_xor(s2, o, 32);
    }
    __shared__ float ps[16], ps2[16];
    const int wid = threadIdx.x >> 5;
    const int nw = blockDim.x >> 5;
    if ((threadIdx.x & 31) == 0) { ps[wid] = s; ps2[wid] = s2; }
    __syncthreads();
    if (threadIdx.x < 32) {
        s  = (threadIdx.x < nw) ? ps[threadIdx.x]  : 0.0f;
        s2 = (threadIdx.x < nw) ? ps2[threadIdx.x] : 0.0f;
#pragma unroll
        for (int o = 16; o > 0; o >>= 1) {
            s  += __shfl_xor(s,  o, 32);
            s2 += __shfl_xor(s2, o, 32);
        }
        if (threadIdx.x == 0) { ps[0] = s; ps2[0] = s2; }
    }
    __syncthreads();
    const float mean = ps[0] / (float)D;
    const float var  = ps2[0] / (float)D - mean * mean;
    const float rstd = rsqrtf(var + 1e-5f);
    for (int i = threadIdx.x; i < D; i += blockDim.x) {
        float v = (x[i] - mean) * rstd * g[i] + b[i];
        if (do_relu) v = fmaxf(v, 0.0f);
        y[i] = v;
    }
}

// ---------------------------------------------------------------------------
// causal depthwise conv (k=4, left pad 3) on first D_INNER cols of XZ + SiLU
// ---------------------------------------------------------------------------
__global__ void k_conv4_silu(const float* __restrict__ XZ,
                             const float* __restrict__ cw,
                             const float* __restrict__ cb,
                             float* __restrict__ XIN, int ldxz)
{
    const int idx = blockIdx.x * blockDim.x + threadIdx.x;
    if (idx >= KB * KT * K_DI) return;
    const int d = idx % K_DI;
    const int t = (idx / K_DI) % KT;
    const int bb = idx / (K_DI * KT);
    const float* w = cw + d * 4;
    float s = cb[d];
#pragma unroll
    for (int j = 0; j < 4; ++j) {
        const int tt = t - 3 + j;
        if (tt >= 0) s += XZ[(size_t)(bb * KT + tt) * ldxz + d] * w[j];
    }
    XIN[(size_t)(bb * KT + t) * K_DI + d] = silu_f(s);
}

// ---------------------------------------------------------------------------
// selective scan: one thread per (batch, d_inner) channel, 16 states in regs
// fuses: dA=exp(dt*A), state update, C contraction, +xin*D, *silu(z)
// ---------------------------------------------------------------------------
__global__ void k_scan(const float* __restrict__ PROJ,   // [BT,64] (B=32..47,C=48..63)
                       const float* __restrict__ DT,     // [BT,1024]
                       const float* __restrict__ XIN,    // [BT,1024]
                       const float* __restrict__ XZ,     // [BT,2048] (z = cols 1024..)
                       const float* __restrict__ A_log,  // [1024,16]
                       const float* __restrict__ Dp,     // [1024]
                       float* __restrict__ Y)            // [BT,1024]
{
    const int idx = blockIdx.x * blockDim.x + threadIdx.x;
    if (idx >= KB * K_DI) return;
    const int d = idx % K_DI;
    const int bb = idx / K_DI;

    float Av[K_DS];
#pragma unroll
    for (int s = 0; s < K_DS; ++s) Av[s] = -expf(A_log[d * K_DS + s]);
    const float Dv = Dp[d];

    float hs[K_DS];
#pragma unroll
    for (int s = 0; s < K_DS; ++s) hs[s] = 0.0f;

    for (int t = 0; t < KT; ++t) {
        const size_t row = (size_t)bb * KT + t;
        const float dtv = DT[row * K_DI + d];
        const float xv  = XIN[row * K_DI + d];
        const float dtx = dtv * xv;
        const float* pr = PROJ + row * (K_DTR + 2 * K_DS);
        float acc = 0.0f;
#pragma unroll
        for (int s = 0; s < K_DS; ++s) {
            const float dA = __expf(dtv * Av[s]);
            hs[s] = dA * hs[s] + dtx * pr[K_DTR + s];
            acc += hs[s] * pr[K_DTR + K_DS + s];
        }
        float yv = acc + xv * Dv;
        const float zv = XZ[row * (2 * K_DI) + K_DI + d];
        yv *= silu_f(zv);
        Y[row * K_DI + d] = yv;
    }
}

// ---------------------------------------------------------------------------
// depthwise conv k=31 (same pad 15) on LN2 output, then h = a + conv
// ---------------------------------------------------------------------------
__global__ void k_conv31_add(const float* __restrict__ XN,
                             const float* __restrict__ w,
                             const float* __restrict__ cb,
                             const float* __restrict__ Aa,
                             float* __restrict__ H)
{
    const int idx = blockIdx.x * blockDim.x + threadIdx.x;
    if (idx >= KB * KT * K_DM) return;
    const int d = idx % K_DM;
    const int t = (idx / K_DM) % KT;
    const int bb = idx / (K_DM * KT);
    const float* wd = w + d * 31;
    float s = cb[d];
#pragma unroll 4
    for (int j = 0; j < 31; ++j) {
        const int tt = t - 15 + j;
        if (tt >= 0 && tt < KT) s += XN[(size_t)(bb * KT + tt) * K_DM + d] * wd[j];
    }
    const size_t row = (size_t)(bb * KT + t);
    H[row * K_DM + d] = Aa[row * K_DM + d] + s;
}

// ---------------------------------------------------------------------------
__global__ void k_mask(const int* __restrict__ lengths, float* __restrict__ out)
{
    const int idx = blockIdx.x * blockDim.x + threadIdx.x;
    if (idx >= KB * KT) return;
    const int t = idx % KT;
    const int bb = idx / KT;
    out[idx] = (t < lengths[bb]) ? 1.0f : 0.0f;
}

// ---------------------------------------------------------------------------
static inline void launch_gemm(hipStream_t s, const float* A, const float* W,
                               const float* bias, const float* resid, float* C,
                               int M, int N, int K, int lda, int ldw, int ldc,
                               int flags)
{
    const int tiles = ((M + 15) / 16) * ((N + 63) / 64);
    const int wpb = 8;                           // 8 waves / block (256 thr)
    dim3 blk(256);
    dim3 grd((tiles + wpb - 1) / wpb);
    k_gemm_wmma_f32<<<grd, blk, 0, s>>>(A, W, bias, resid, C,
                                        M, N, K, lda, ldw, ldc, flags);
}

extern "C" void kernel_launch(void* const* d_in, const int* in_sizes, int n_in,
                              void* d_out, int out_size, void* d_ws, size_t ws_size,
                              hipStream_t stream)
{
    const float* x        = (const float*)d_in[0];
    const int*   lengths  = (const int*)d_in[1];
    const float* w_in0    = (const float*)d_in[2];
    const float* ln_in_g  = (const float*)d_in[3];
    const float* ln_in_b  = (const float*)d_in[4];
    const float* ln1_g    = (const float*)d_in[5];
    const float* ln1_b    = (const float*)d_in[6];
    const float* W_inproj = (const float*)d_in[7];
    const float* conv_w   = (const float*)d_in[8];
    const float* conv_b   = (const float*)d_in[9];
    const float* W_xproj  = (const float*)d_in[10];
    const float* W_dt     = (const float*)d_in[11];
    const float* b_dt     = (const float*)d_in[12];
    const float* A_log    = (const float*)d_in[13];
    const float* Dp       = (const float*)d_in[14];
    const float* W_mo     = (const float*)d_in[15];
    const float* ln2_g    = (const float*)d_in[16];
    const float* ln2_b    = (const float*)d_in[17];
    const float* dc_w     = (const float*)d_in[18];
    const float* dc_b     = (const float*)d_in[19];
    const float* w_out    = (const float*)d_in[20];
    const float* b_out    = (const float*)d_in[21];
    float* out = (float*)d_out;

    // workspace carve-up (floats)
    float* ws = (float*)d_ws;
    float* buf_h    = ws; ws += (size_t)K_BT * K_DM;
    float* buf_a    = ws; ws += (size_t)K_BT * K_DM;
    float* buf_xn   = ws; ws += (size_t)K_BT * K_DM;
    float* buf_xz   = ws; ws += (size_t)K_BT * 2 * K_DI;
    float* buf_xin  = ws; ws += (size_t)K_BT * K_DI;
    float* buf_proj = ws; ws += (size_t)K_BT * (K_DTR + 2 * K_DS);
    float* buf_dt   = ws; ws += (size_t)K_BT * K_DI;
    float* buf_y    = ws; ws += (size_t)K_BT * K_DI;

    // ---- stem: h = relu(LN(x @ w_in0)) ----
    launch_gemm(stream, x, w_in0, nullptr, nullptr, buf_a,
                K_BT, K_DM, K_IN, K_IN, K_DM, K_DM, 0);
    k_layernorm<<<K_BT, 256, 0, stream>>>(buf_a, ln_in_g, ln_in_b, buf_h, K_DM, 1);

    for (int i = 0; i < K_NL; ++i) {
        // LN1
        k_layernorm<<<K_BT, 256, 0, stream>>>(buf_h, ln1_g + i * K_DM,
                                              ln1_b + i * K_DM, buf_xn, K_DM, 0);
        // xz = xn @ W_inproj  (2000x512 @ 512x2048)
        launch_gemm(stream, buf_xn, W_inproj + (size_t)i * K_DM * 2 * K_DI,
                    nullptr, nullptr, buf_xz,
                    K_BT, 2 * K_DI, K_DM, K_DM, 2 * K_DI, 2 * K_DI, 0);
        // xin = silu(causal dwconv4(xz[:, :1024]))
        {
            const int tot = KB * KT * K_DI;
            k_conv4_silu<<<(tot + 255) / 256, 256, 0, stream>>>(
                buf_xz, conv_w + (size_t)i * K_DI * 4, conv_b + (size_t)i * K_DI,
                buf_xin, 2 * K_DI);
        }
        // proj = xin @ W_xproj  (N = 64)
        launch_gemm(stream, buf_xin, W_xproj + (size_t)i * K_DI * (K_DTR + 2 * K_DS),
                    nullptr, nullptr, buf_proj,
                    K_BT, K_DTR + 2 * K_DS, K_DI, K_DI,
                    K_DTR + 2 * K_DS, K_DTR + 2 * K_DS, 0);
        // dt = softplus(proj[:, :32] @ W_dt + b_dt)
        launch_gemm(stream, buf_proj, W_dt + (size_t)i * K_DTR * K_DI,
                    b_dt + (size_t)i * K_DI, nullptr, buf_dt,
                    K_BT, K_DI, K_DTR, K_DTR + 2 * K_DS, K_DI, K_DI,
                    FLAG_BIAS | FLAG_SOFTPLUS);
        // selective scan + D skip + silu(z) gate
        {
            const int tot = KB * K_DI;
            k_scan<<<(tot + 255) / 256, 256, 0, stream>>>(
                buf_proj, buf_dt, buf_xin, buf_xz,
                A_log + (size_t)i * K_DI * K_DS, Dp + (size_t)i * K_DI, buf_y);
        }
        // a = h + y @ W_mo
        launch_gemm(stream, buf_y, W_mo + (size_t)i * K_DI * K_DM,
                    nullptr, buf_h, buf_a,
                    K_BT, K_DM, K_DI, K_DI, K_DM, K_DM, FLAG_RESID);
        // LN2
        k_layernorm<<<K_BT, 256, 0, stream>>>(buf_a, ln2_g + i * K_DM,
                                              ln2_b + i * K_DM, buf_xn, K_DM, 0);
        // h = a + dwconv31(LN2(a))
        {
            const int tot = KB * KT * K_DM;
            k_conv31_add<<<(tot + 255) / 256, 256, 0, stream>>>(
                buf_xn, dc_w + (size_t)i * K_DM * 31, dc_b + (size_t)i * K_DM,
                buf_a, buf_h);
        }
    }

    // ---- head: out = h @ w_out + b_out ----
    launch_gemm(stream, buf_h, w_out, b_out, nullptr, out,
                K_BT, K_OUT, K_DM, K_DM, K_OUT, K_OUT, FLAG_BIAS);
    // seq mask appended after out_flat
    k_mask<<<(KB * KT + 255) / 256, 256, 0, stream>>>(lengths, out + (size_t)K_BT * K_OUT);
}